// SegmentAwarePool_35064113004891
// MI455X (gfx1250) — compile-verified
//
#include <hip/hip_runtime.h>

#define SEP_ID 102
#define B_SZ 64
#define S_SZ 512
#define H_SZ 1024

typedef __attribute__((ext_vector_type(2))) float v2f;
typedef __attribute__((ext_vector_type(8))) float v8f;

struct SegInfo {
  int fs;        // first_sep (0 => title empty); gated so no-sep => 0
  int le;        // lead end (exclusive); gated so no-sep => 0
  float tscale;  // 1/title_cnt or 0
  float lscale;  // 1/lead_cnt or 0
};

// Phase 1: per-batch SEP statistics. 64 threads total; negligible cost.
__global__ void seg_stats_kernel(const int* __restrict__ ids,
                                 SegInfo* __restrict__ info) {
  int b = blockIdx.x * blockDim.x + threadIdx.x;
  if (b >= B_SZ) return;
  const int* row = ids + b * S_SZ;
  int n = 0, first = -1, last = -1;
  for (int s = 0; s < S_SZ; ++s) {
    int v = row[s];
    if (v == SEP_ID) {
      ++n;
      if (first < 0) first = s;
      last = s;
    }
  }
  int fs, le;
  if (n == 0) {
    fs = 0;  // title mask: pos in [1, 0) -> empty
    le = 0;  // lead mask: pos in (0, 0) -> empty
  } else {
    fs = first;
    le = (n >= 2) ? last : S_SZ;  // exclusive upper bound of lead segment
  }
  int tcnt = fs - 1;       // |[1, fs)|
  if (tcnt < 0) tcnt = 0;
  int lcnt = le - fs - 1;  // |(fs, le)|
  if (lcnt < 0) lcnt = 0;
  SegInfo o;
  o.fs = fs;
  o.le = le;
  o.tscale = (tcnt > 0) ? (1.0f / (float)tcnt) : 0.0f;
  o.lscale = (lcnt > 0) ? (1.0f / (float)lcnt) : 0.0f;
  info[b] = o;
}

// Phase 2: WMMA masked reduction. Each wave owns one (batch, 16-column) tile
// and sweeps K=S in steps of 4 with V_WMMA_F32_16X16X4_F32.
//   A (16x4): row 0 = title mask, row 1 = lead mask, rows 2..15 = 0.
//   B (4x16): hidden_states[b][k0..k0+3][n0..n0+15].
// A layout: lane<16 -> M=lane, K={0,1}; lane>=16 -> M=lane-16, K={2,3}.
// B layout: lane<16 -> N=lane,  VGPR0=K0/VGPR1=K1; lane>=16 -> K2/K3.
// C/D layout: VGPR r, lanes 0..15 => (M=r, N=lane). Rows 0,1 = the two sums.
//
// Mask construction is branchless: per lane the mask is "p in [lo, lo+len)"
// with loop-invariant lo/len, so each A element is ONE unsigned compare plus
// a cndmask -- no EXEC divergence in the hot loop.
__global__ void __launch_bounds__(256)
seg_pool_wmma_kernel(const float* __restrict__ hs,
                     const SegInfo* __restrict__ info,
                     float* __restrict__ out) {
  const int lane = threadIdx.x & 31;
  const int wave = threadIdx.x >> 5;                 // 0..7
  const int b = blockIdx.y;
  const int n0 = blockIdx.x * 128 + wave * 16;       // column tile base

  const SegInfo si = info[b];                        // scalar (uniform) load
  const int m = lane & 15;                           // A row / B column index
  const int khalf = (lane >> 4) << 1;                // 0 or 2

  // Per-lane loop-invariant mask window [lo, lo+len).
  int lo = 0;
  int len = 0;
  if (m == 0) {            // title: [1, fs)
    lo = 1;
    len = si.fs - 1;
  } else if (m == 1) {     // lead: (fs, le)
    lo = si.fs + 1;
    len = si.le - si.fs - 1;
  }
  if (len < 0) len = 0;    // empty / no-SEP -> exact-zero mask
  const unsigned ulen = (unsigned)len;

  const float* __restrict__ base =
      hs + (size_t)b * (S_SZ * H_SZ) + (size_t)n0 + (size_t)m;

  v8f c = {0.f, 0.f, 0.f, 0.f, 0.f, 0.f, 0.f, 0.f};

  // rel = p0 - lo for this lane's first K element; advances by 4 per step.
  int rel = khalf - lo;

#pragma unroll 8
  for (int k0 = 0; k0 < S_SZ; k0 += 4) {
    // Branchless A-matrix mask values (one cmp + cndmask each).
    v2f a;
    a.x = ((unsigned)rel < ulen) ? 1.0f : 0.0f;
    a.y = ((unsigned)(rel + 1) < ulen) ? 1.0f : 0.0f;

    // B-matrix tile loads (coalesced 64B per half-wave per row).
    const float* p = base + (size_t)(k0 + khalf) * H_SZ;
    v2f bm;
    bm.x = p[0];
    bm.y = p[H_SZ];

    // HBM streaming hint (speculative; OOB tail prefetch is dropped).
    __builtin_prefetch(p + 16 * H_SZ, 0, 0);

    // D = A x B + C  (accumulate across K)
    c = __builtin_amdgcn_wmma_f32_16x16x4_f32(
        /*neg_a=*/false, a, /*neg_b=*/false, bm,
        /*c_mod=*/(short)0, c, /*reuse_a=*/false, /*reuse_b=*/false);

    rel += 4;
  }

  // Row 0 (title) and row 1 (lead) live in c[0]/c[1] of lanes 0..15.
  if (lane < 16) {
    const int col = n0 + lane;
    out[(size_t)b * H_SZ + col] = c[0] * si.tscale;
    out[(size_t)(B_SZ * H_SZ) + (size_t)b * H_SZ + col] = c[1] * si.lscale;
  }
}

extern "C" void kernel_launch(void* const* d_in, const int* in_sizes, int n_in,
                              void* d_out, int out_size, void* d_ws,
                              size_t ws_size, hipStream_t stream) {
  (void)in_sizes;
  (void)n_in;
  (void)out_size;
  (void)ws_size;

  const float* hidden = (const float*)d_in[0];  // [B, S, H] f32
  const int* ids = (const int*)d_in[1];         // [B, S] int
  float* out = (float*)d_out;                   // [2, B, H] f32 (title, lead)
  SegInfo* info = (SegInfo*)d_ws;               // 64 * 16 bytes

  seg_stats_kernel<<<1, 64, 0, stream>>>(ids, info);

  dim3 grid(H_SZ / 128, B_SZ);  // 8 x 64 blocks
  dim3 block(256);              // 8 waves per block, one 16-col tile per wave
  seg_pool_wmma_kernel<<<grid, block, 0, stream>>>(hidden, info, out);
}